// RWRKernelAttention_23974507446666
// MI455X (gfx1250) — compile-verified
//
#include <hip/hip_runtime.h>
#include <cmath>

// ---------------- problem constants (from reference setup_inputs) ------------
#define BB 2
#define HH 12
#define NN 1024
#define DD 64
#define BHN (BB*HH)            // 24
#define TOPKK 32
#define WIN 128
#define NEGV (-1e30f)
#define QK_SCALE 0.125f        // 64^-0.5
#define LENSV 0.3f

typedef float v2f __attribute__((ext_vector_type(2)));
typedef float v8f __attribute__((ext_vector_type(8)));

__device__ __forceinline__ v8f wmma_f32_4(v2f a, v2f b, v8f c) {
  // (neg_a, A, neg_b, B, c_mod, C, reuse_a, reuse_b)
  return __builtin_amdgcn_wmma_f32_16x16x4_f32(false, a, false, b, (short)0, c,
                                               false, false);
}

// ============================================================================
// Kernel 1: S = q @ k^T * scale    (per bh; A=q[1024x64], B^T=k[1024x64])
// grid (16, 8, 24), block 256 (8 waves). Wave -> 16x64 strip of S.
// ============================================================================
__global__ void scores_wmma_kernel(const float* __restrict__ q,
                                   const float* __restrict__ k,
                                   float* __restrict__ S) {
  const int bh   = blockIdx.z;
  const int tid  = threadIdx.x;
  const int wave = tid >> 5;
  const int lane = tid & 31;
  const int half = lane >> 4;
  const int lid  = lane & 15;
  const int rt   = blockIdx.y * 8 + wave;   // row tile 0..63
  const int c0   = blockIdx.x * 64;         // col base

  const float* Q = q + (size_t)bh * NN * DD;
  const float* K = k + (size_t)bh * NN * DD;
  float* Sp = S + (size_t)bh * NN * NN;

  const int m = rt * 16 + lid;
  v8f acc0 = {}, acc1 = {}, acc2 = {}, acc3 = {};
#pragma unroll
  for (int kk = 0; kk < DD; kk += 4) {
    const int kc = kk + 2 * half;           // this half-wave's K pair
    v2f a  = *(const v2f*)(Q + (size_t)m * DD + kc);
    v2f b0 = *(const v2f*)(K + (size_t)(c0 +  0 + lid) * DD + kc);
    v2f b1 = *(const v2f*)(K + (size_t)(c0 + 16 + lid) * DD + kc);
    v2f b2 = *(const v2f*)(K + (size_t)(c0 + 32 + lid) * DD + kc);
    v2f b3 = *(const v2f*)(K + (size_t)(c0 + 48 + lid) * DD + kc);
    acc0 = wmma_f32_4(a, b0, acc0);
    acc1 = wmma_f32_4(a, b1, acc1);
    acc2 = wmma_f32_4(a, b2, acc2);
    acc3 = wmma_f32_4(a, b3, acc3);
  }
#pragma unroll
  for (int p = 0; p < 8; ++p) {
    const int row = rt * 16 + p + 8 * half;
    float* r = Sp + (size_t)row * NN + c0 + lid;
    r[0]  = acc0[p] * QK_SCALE;
    r[16] = acc1[p] * QK_SCALE;
    r[32] = acc2[p] * QK_SCALE;
    r[48] = acc3[p] * QK_SCALE;
  }
}

// ============================================================================
// Kernel 2: banded softmax + y_local -> out.  block 256 = 4 rows x 64 dims.
// grid (N/4, BH)
// ============================================================================
__global__ void ylocal_kernel(const float* __restrict__ S,
                              const float* __restrict__ v,
                              float* __restrict__ out) {
  __shared__ float pbuf[4][2 * WIN + 1];
  __shared__ float red[4][64];
  const int tid = threadIdx.x;
  const int ty  = tid >> 6;       // row in block 0..3
  const int d   = tid & 63;       // feature dim
  const int bh  = blockIdx.y;
  const int i   = blockIdx.x * 4 + ty;

  const int j0 = (i - WIN > 0) ? i - WIN : 0;
  const int j1 = (i + WIN < NN - 1) ? i + WIN : NN - 1;
  const int L  = j1 - j0 + 1;

  const float* srow = S + ((size_t)bh * NN + i) * NN;

  // row max over band
  float m = -3.4e38f;
  for (int j = j0 + d; j <= j1; j += 64) m = fmaxf(m, srow[j]);
  red[ty][d] = m;
  __syncthreads();
#pragma unroll
  for (int s = 32; s > 0; s >>= 1) {
    if (d < s) red[ty][d] = fmaxf(red[ty][d], red[ty][d + s]);
    __syncthreads();
  }
  m = red[ty][0];
  __syncthreads();

  // exp + sum
  float sum = 0.f;
  for (int j = j0 + d; j <= j1; j += 64) {
    float e = expf(srow[j] - m);
    pbuf[ty][j - j0] = e;
    sum += e;
  }
  red[ty][d] = sum;
  __syncthreads();
#pragma unroll
  for (int s = 32; s > 0; s >>= 1) {
    if (d < s) red[ty][d] += red[ty][d + s];
    __syncthreads();
  }
  const float inv = 1.f / red[ty][0];
  __syncthreads();

  // y_local[i, d] = sum_j p_j * v[j, d]
  const float* V = v + (size_t)bh * NN * DD;
  float acc = 0.f;
  for (int jj = 0; jj < L; ++jj)
    acc = fmaf(pbuf[ty][jj], V[(size_t)(j0 + jj) * DD + d], acc);
  out[((size_t)bh * NN + i) * DD + d] = acc * inv;
}

// ============================================================================
// Kernel 3: per-row top-32 (optionally excluding local window) -> vals, idx
// grid (N, BH), block 256. Tie-break: smaller index (matches lax.top_k).
// ============================================================================
__global__ void topk32_kernel(const float* __restrict__ M,
                              float* __restrict__ ovals,
                              int* __restrict__ oidx,
                              int maskLocal) {
  __shared__ float data[NN];
  __shared__ float rv[256];
  __shared__ int   ri[256];
  const int i   = blockIdx.x;
  const int bh  = blockIdx.y;
  const int tid = threadIdx.x;
  const float* row = M + ((size_t)bh * NN + i) * NN;

  for (int j = tid; j < NN; j += 256) {
    float x = row[j];
    if (maskLocal) {
      int dd = (j > i) ? (j - i) : (i - j);
      if (dd <= WIN) x = NEGV;
    }
    data[j] = x;
  }
  __syncthreads();

  for (int t = 0; t < TOPKK; ++t) {
    float bv = -3.4e38f;
    int bi = -1;
    for (int j = tid; j < NN; j += 256) {
      float x = data[j];
      if (x > bv) { bv = x; bi = j; }    // strict > keeps smallest index
    }
    rv[tid] = bv; ri[tid] = bi;
    __syncthreads();
#pragma unroll
    for (int s = 128; s > 0; s >>= 1) {
      if (tid < s) {
        float v2 = rv[tid + s]; int i2 = ri[tid + s];
        if (v2 > rv[tid] || (v2 == rv[tid] && i2 < ri[tid])) {
          rv[tid] = v2; ri[tid] = i2;
        }
      }
      __syncthreads();
    }
    if (tid == 0) {
      ovals[((size_t)bh * NN + i) * TOPKK + t] = rv[0];
      oidx [((size_t)bh * NN + i) * TOPKK + t] = ri[0];
      data[ri[0]] = -3.4e38f;
    }
    __syncthreads();
  }
}

// ============================================================================
// Kernel 4: scatter normalized thresholded top-k into dense P (P pre-zeroed)
// one thread per row
// ============================================================================
__global__ void scatterP_kernel(const float* __restrict__ tkv,
                                const int* __restrict__ tki,
                                float* __restrict__ P) {
  const int r = blockIdx.x * 256 + threadIdx.x;
  if (r >= BHN * NN) return;
  const int bh = r / NN, i = r % NN;
  const float* wv = tkv + (size_t)r * TOPKK;
  const int*   wi = tki + (size_t)r * TOPKK;
  float w[TOPKK];
  float s = 0.f;
#pragma unroll
  for (int k = 0; k < TOPKK; ++k) {
    float x = wv[k];
    w[k] = (x > 0.0f) ? x : 0.f;     // THRESH = 0.0
    s += w[k];
  }
  const float inv = 1.f / (s + 1e-9f);
  float* prow = P + ((size_t)bh * NN + i) * NN;
#pragma unroll
  for (int k = 0; k < TOPKK; ++k)
    if (w[k] != 0.f) prow[wi[k]] = w[k] * inv;   // indices distinct in a row
}

// ============================================================================
// Kernel 5: Racc = 0.2*I + 0.16*P   (alpha*I + alpha*(1-alpha)*P, R1 = P)
// grid (N*N/256, BH)
// ============================================================================
__global__ void racc_init_kernel(const float* __restrict__ P,
                                 float* __restrict__ Racc) {
  const int idx = blockIdx.x * 256 + threadIdx.x;   // index within plane
  const int bh  = blockIdx.y;
  const int i = idx >> 10, j = idx & (NN - 1);
  const size_t o = (size_t)bh * NN * NN + idx;
  Racc[o] = 0.16f * P[o] + ((i == j) ? 0.2f : 0.f);
}

// ============================================================================
// Kernel 6: D = A @ B (NxN, f32 WMMA) ; Racc += coef * D
// grid (16, 8, 24), block 256 (8 waves). Block -> 128 rows x 64 cols.
// B panel (64 K x 64 cols) staged transposed in LDS once per block: all 8
// waves share it; fragment pair {B[k][c],B[k+1][c]} is one aligned ds_load_b64
// (stride 66 floats keeps 8B alignment and spreads banks).
// ============================================================================
#define KC 64
#define BS_STRIDE 66
__global__ void gemm_racc_kernel(const float* __restrict__ A,
                                 const float* __restrict__ Bm,
                                 float* __restrict__ Dout,
                                 float* __restrict__ Racc,
                                 float coef) {
  __shared__ float Bs[64 * BS_STRIDE];   // Bs[c*66 + k], transposed panel
  const int bh   = blockIdx.z;
  const int tid  = threadIdx.x;
  const int wave = tid >> 5;
  const int lane = tid & 31;
  const int half = lane >> 4;
  const int lid  = lane & 15;
  const int rt   = blockIdx.y * 8 + wave;
  const int c0   = blockIdx.x * 64;

  const float* Ap = A    + (size_t)bh * NN * NN;
  const float* Bp = Bm   + (size_t)bh * NN * NN;
  float* Dp       = Dout + (size_t)bh * NN * NN;
  float* Rp       = Racc + (size_t)bh * NN * NN;

  const int m = rt * 16 + lid;
  const float* Arow = Ap + (size_t)m * NN;

  v8f acc0 = {}, acc1 = {}, acc2 = {}, acc3 = {};
  for (int kk0 = 0; kk0 < NN; kk0 += KC) {
    __syncthreads();
    // cooperative staging: B[kk0..kk0+63][c0..c0+63] -> Bs[c][k] (transposed)
#pragma unroll
    for (int r = 0; r < (KC * 64) / 256; ++r) {
      const int idx = tid + r * 256;
      const int k = idx >> 6;          // 0..63
      const int c = idx & 63;          // coalesced inner dim
      Bs[c * BS_STRIDE + k] = Bp[(size_t)(kk0 + k) * NN + c0 + c];
    }
    __syncthreads();
#pragma unroll 4
    for (int kk = 0; kk < KC; kk += 4) {
      const int kc = kk + 2 * half;
      v2f a  = *(const v2f*)(Arow + kk0 + kc);
      v2f b0 = *(const v2f*)(&Bs[(lid +  0) * BS_STRIDE + kc]);
      v2f b1 = *(const v2f*)(&Bs[(lid + 16) * BS_STRIDE + kc]);
      v2f b2 = *(const v2f*)(&Bs[(lid + 32) * BS_STRIDE + kc]);
      v2f b3 = *(const v2f*)(&Bs[(lid + 48) * BS_STRIDE + kc]);
      acc0 = wmma_f32_4(a, b0, acc0);
      acc1 = wmma_f32_4(a, b1, acc1);
      acc2 = wmma_f32_4(a, b2, acc2);
      acc3 = wmma_f32_4(a, b3, acc3);
    }
  }
#pragma unroll
  for (int p = 0; p < 8; ++p) {
    const int row = rt * 16 + p + 8 * half;
    const size_t o = (size_t)row * NN + c0 + lid;
    Dp[o]      = acc0[p];
    Dp[o + 16] = acc1[p];
    Dp[o + 32] = acc2[p];
    Dp[o + 48] = acc3[p];
    Rp[o]      += coef * acc0[p];
    Rp[o + 16] += coef * acc1[p];
    Rp[o + 32] += coef * acc2[p];
    Rp[o + 48] += coef * acc3[p];
  }
}

// ============================================================================
// Kernel 7: out += LENS * sum_k rvals[k] * v[ridx[k]]   (block = 4 rows x 64d)
// ============================================================================
__global__ void combine_kernel(const float* __restrict__ v,
                               const float* __restrict__ rvals,
                               const int* __restrict__ ridx,
                               float* __restrict__ out) {
  const int tid = threadIdx.x;
  const int ty  = tid >> 6;
  const int d   = tid & 63;
  const int bh  = blockIdx.y;
  const int i   = blockIdx.x * 4 + ty;
  const float* V = v + (size_t)bh * NN * DD;
  const size_t rb = ((size_t)bh * NN + i) * TOPKK;
  float acc = 0.f;
#pragma unroll
  for (int k = 0; k < TOPKK; ++k)
    acc = fmaf(rvals[rb + k], V[(size_t)ridx[rb + k] * DD + d], acc);
  const size_t o = ((size_t)bh * NN + i) * DD + d;
  out[o] = out[o] + LENSV * acc;
}

// ============================================================================
extern "C" void kernel_launch(void* const* d_in, const int* in_sizes, int n_in,
                              void* d_out, int out_size, void* d_ws,
                              size_t ws_size, hipStream_t stream) {
  (void)in_sizes; (void)n_in; (void)out_size; (void)ws_size;
  const float* q = (const float*)d_in[0];
  const float* k = (const float*)d_in[1];
  const float* v = (const float*)d_in[2];
  float* out = (float*)d_out;

  const size_t PLANE = (size_t)BHN * NN * NN;          // 25165824 floats
  float* ws   = (float*)d_ws;
  float* S    = ws;                  // scores, later recycled as R (odd steps)
  float* P    = ws + PLANE;
  float* Rb   = ws + 2 * PLANE;      // R (even steps)
  float* Racc = ws + 3 * PLANE;
  float* tkv  = ws + 4 * PLANE;                        // BH*N*32 floats
  int*   tki  = (int*)(tkv + (size_t)BHN * NN * TOPKK);
  float* tkv2 = (float*)(tki + (size_t)BHN * NN * TOPKK);
  int*   tki2 = (int*)(tkv2 + (size_t)BHN * NN * TOPKK);

  // 1. S = q k^T * scale (WMMA f32)
  scores_wmma_kernel<<<dim3(16, 8, BHN), 256, 0, stream>>>(q, k, S);
  // 2. banded softmax + y_local -> out
  ylocal_kernel<<<dim3(NN / 4, BHN), 256, 0, stream>>>(S, v, out);
  // 3. top-32 of non-local scores
  topk32_kernel<<<dim3(NN, BHN), 256, 0, stream>>>(S, tkv, tki, 1);
  // 4. build dense P
  hipMemsetAsync(P, 0, PLANE * sizeof(float), stream);
  scatterP_kernel<<<dim3((BHN * NN + 255) / 256), 256, 0, stream>>>(tkv, tki, P);
  // 5. Racc = 0.2 I + 0.16 P   (R1 = P handled implicitly)
  racc_init_kernel<<<dim3(NN * NN / 256, BHN), 256, 0, stream>>>(P, Racc);
  // 6. R2 = P@P, R3 = R2@P, R4 = R3@P with fused Racc += c_t * R_t
  gemm_racc_kernel<<<dim3(16, 8, BHN), 256, 0, stream>>>(P,  P, Rb, Racc, 0.128f);
  gemm_racc_kernel<<<dim3(16, 8, BHN), 256, 0, stream>>>(Rb, P, S,  Racc, 0.1024f);
  gemm_racc_kernel<<<dim3(16, 8, BHN), 256, 0, stream>>>(S,  P, Rb, Racc, 0.08192f);
  // 7. top-32 of Racc rows
  topk32_kernel<<<dim3(NN, BHN), 256, 0, stream>>>(Racc, tkv2, tki2, 0);
  // 8. out += LENS * y_rwr
  combine_kernel<<<dim3(NN / 4, BHN), 256, 0, stream>>>(v, tkv2, tki2, out);
}